// RNN_parallel_5093831213142
// MI455X (gfx1250) — compile-verified
//
#include <hip/hip_runtime.h>
#include <hip/hip_bf16.h>

// 13 scalar-hidden RNNs + Linear(13->1), single timestep, h0 = 0.
// ~600 FLOPs / ~1.4 KB -> pure latency problem: one wave32, one workgroup.
// 13x20 fp32 matvec = 5 chained V_WMMA_F32_16X16X4_F32 (exact fp32 FMA/RNE).
// Branchless throughout (clamped addresses + selects + range-stable tanh);
// final reduction uses ds_swizzle_b32 xor-butterflies (1 instr per step).

typedef __attribute__((ext_vector_type(2))) float v2f;
typedef __attribute__((ext_vector_type(4))) float v4f;
typedef __attribute__((ext_vector_type(8))) float v8f;

#define N_RNN 13
#define D_IN  20

// group-of-32 ds_swizzle pattern: and_mask=0x1f, or_mask=0, xor_mask=x
#define SWIZ_XOR(x) (((x) << 10) | 0x1F)

__device__ __forceinline__ float swizzle_xor_add(float v, int imm_off) {
    return v;  // unused fallback (never called)
}

__launch_bounds__(32)
__global__ void rnn13_wmma_f32_kernel(const float* __restrict__ input_ids,
                                      const float* __restrict__ W_ih,
                                      const float* __restrict__ b_ih,
                                      const float* __restrict__ W_hh,
                                      const float* __restrict__ b_hh,
                                      const float* __restrict__ W_out,
                                      const float* __restrict__ b_out,
                                      float* __restrict__ out) {
    __shared__ float xs[32];                      // gathered x, zero padded
    __shared__ __align__(16) float dslab[32 * 8]; // per-lane D spill (32B/lane)

    const int lane = threadIdx.x;                 // 0..31, exactly one wave32

    // ---- branchless gather of x (drop features 10,11) -------------------
    int gidx = (lane < 10) ? lane : (lane + 2);
    gidx = (gidx > 21) ? 21 : gidx;               // clamp -> always in bounds
    const float xval = input_ids[gidx];
    xs[lane] = (lane < D_IN) ? xval : 0.0f;       // select, no predicated load
    __syncthreads();

    // ---- A-matrix 16x4 f32 layout (ISA 7.12.2) --------------------------
    //   lanes 0-15 : row m=lane,    VGPR0=K0, VGPR1=K1
    //   lanes 16-31: row m=lane-16, VGPR0=K2, VGPR1=K3
    const int   m       = lane & 15;
    const int   khalf   = (lane >> 4) << 1;               // 0 or 2
    const int   row     = (m < N_RNN) ? m : (N_RNN - 1);  // clamped address
    const float rowmask = (m < N_RNN) ? 1.0f : 0.0f;      // zero rows 13..15
    const float* arow   = W_ih + row * D_IN + khalf;      // 8B-aligned pairs

    v8f acc = {};
#pragma unroll
    for (int c = 0; c < 5; ++c) {
        v2f a = *(const v2f*)(arow + 4 * c);      // global_load_b64, clause-able
        a.x *= rowmask;
        a.y *= rowmask;
        v2f b;                                    // B row k broadcast over cols
        b.x = xs[4 * c + khalf];
        b.y = xs[4 * c + khalf + 1];
        // (neg_a, A, neg_b, B, c_mod, C, reuse_a, reuse_b)
        acc = __builtin_amdgcn_wmma_f32_16x16x4_f32(
                  false, a, false, b, (short)0, acc, false, false);
    }

    // ---- branchless D spill: every lane stores its 8 accum values -------
    // Lane L holds column L%16, rows 0..7 (L<16) or rows 8..15 (L>=16).
    {
        float* dst = dslab + lane * 8;
        v4f lo = { acc[0], acc[1], acc[2], acc[3] };
        v4f hi = { acc[4], acc[5], acc[6], acc[7] };
        *(v4f*)(dst)     = lo;                    // ds_store_b128
        *(v4f*)(dst + 4) = hi;                    // ds_store_b128
    }
    __syncthreads();

    // ---- parallel epilogue: branchless tanh across lanes ----------------
    // Column 0 of D: row i<8 lives in lane 0's slab, row i>=8 in lane 16's.
    // h0 == 0 so the W_hh*h0 recurrent term vanishes exactly.
    const int   i    = (lane < N_RNN) ? lane : (N_RNN - 1);   // clamped loads
    const int   slot = ((i < 8) ? 0 : (16 * 8 - 8)) + i;      // slab index
    const float z    = dslab[slot] + b_ih[i] + b_hh[i];

    // tanh(z) = 1 - 2/(exp(2z)+1); exp via hardware exp2, rcp via v_rcp_f32.
    // Saturates exactly to +/-1 for large |z| (inf -> 1, 0 -> -1): no branches.
    const float t  = __builtin_amdgcn_exp2f(z * 2.885390081777927f); // 2*log2(e)
    const float th = 1.0f - 2.0f * __builtin_amdgcn_rcpf(t + 1.0f);

    float contrib = W_out[i] * th;
    contrib = (lane < N_RNN) ? contrib : 0.0f;    // mask pad lanes (13..31)

    // ---- xor-butterfly reduction via ds_swizzle_b32 ---------------------
    // Lanes 16..31 hold exact zeros, so offsets 8,4,2,1 (within the low 16
    // lanes) are sufficient: lane 0 ends with the sum of lanes 0..15.
    contrib += __int_as_float(__builtin_amdgcn_ds_swizzle(
                   __float_as_int(contrib), SWIZ_XOR(8)));
    contrib += __int_as_float(__builtin_amdgcn_ds_swizzle(
                   __float_as_int(contrib), SWIZ_XOR(4)));
    contrib += __int_as_float(__builtin_amdgcn_ds_swizzle(
                   __float_as_int(contrib), SWIZ_XOR(2)));
    contrib += __int_as_float(__builtin_amdgcn_ds_swizzle(
                   __float_as_int(contrib), SWIZ_XOR(1)));

    if (lane == 0) out[0] = contrib + b_out[0];
    (void)W_hh;   // unused: h0 = 0
}

extern "C" void kernel_launch(void* const* d_in, const int* in_sizes, int n_in,
                              void* d_out, int out_size, void* d_ws, size_t ws_size,
                              hipStream_t stream) {
    const float* input_ids = (const float*)d_in[0];
    const float* W_ih      = (const float*)d_in[1];
    const float* b_ih      = (const float*)d_in[2];
    const float* W_hh      = (const float*)d_in[3];
    const float* b_hh      = (const float*)d_in[4];
    const float* W_out     = (const float*)d_in[5];
    const float* b_out     = (const float*)d_in[6];
    float* out = (float*)d_out;

    (void)in_sizes; (void)n_in; (void)out_size; (void)d_ws; (void)ws_size;

    rnn13_wmma_f32_kernel<<<1, 32, 0, stream>>>(
        input_ids, W_ih, b_ih, W_hh, b_hh, W_out, b_out, out);
}